// GCN_fusion8_91036126806367
// MI455X (gfx1250) — compile-verified
//
#include <hip/hip_runtime.h>
#include <hip/hip_bf16.h>

// GCN fusion for MI455X (gfx1250, wave32).
// Heavy GEMMs run on the f32 WMMA pipe (v_wmma_f32_16x16x4_f32): exact f32
// numerics on the matrix units. 128x64 block tile, 32x32 per-wave tile,
// B stored transposed in LDS (single b64 per fragment), double-buffered LDS.

typedef __attribute__((ext_vector_type(2))) float v2f;
typedef __attribute__((ext_vector_type(8))) float v8f;

#define MBLK 128
#define NBLK 64
#define KBLK 32
#define ASTR 34   // LDS stride for A rows   (even -> 8B-aligned float2 frags)
#define BTSTR 34  // LDS stride for B^T rows (even -> 8B-aligned float2 frags)

#define WMMA_F32(acc, a, b) \
  acc = __builtin_amdgcn_wmma_f32_16x16x4_f32(false, a, false, b, (short)0, acc, false, false)

// C[M,N] = act(A[M,K] @ B[K,N] (+ bias)), row-major, fuse=1 -> bias+ReLU.
// Requires M%128==0, N%64==0, K%32==0 (true for all calls below).
__launch_bounds__(256)
__global__ void gemm_wmma_f32(const float* __restrict__ A,
                              const float* __restrict__ B,
                              const float* __restrict__ bias,
                              float* __restrict__ C,
                              int M, int N, int K, int fuse) {
  __shared__ __attribute__((aligned(16))) float Alds[2][MBLK * ASTR];   // 2 x 128x32
  __shared__ __attribute__((aligned(16))) float Blds[2][NBLK * BTSTR];  // 2 x (B^T 64x32)

  const int tid  = threadIdx.x;
  const int lane = tid & 31;
  const int wave = tid >> 5;        // 0..7
  const int wm   = wave >> 1;       // 0..3 : 32-row slab
  const int wn   = wave & 1;        // 0..1 : 32-col slab
  const int half = lane >> 4;       // 0/1  : K-pair select / C row-half
  const int l16  = lane & 15;

  const int m0 = blockIdx.y * MBLK;
  const int n0 = blockIdx.x * NBLK;

  v8f acc00 = {}, acc01 = {}, acc10 = {}, acc11 = {};

  // A staging: each thread 4 x float4 (rows rA+32p, cols caA..caA+3 of K-tile)
  const int caA = (tid & 7) * 4;    // 0..28
  const int rA  = tid >> 3;         // 0..31
  // B staging: each thread 2 x float4 (K-rows rB+16p, N-cols cB*4..cB*4+3)
  const int cB = tid & 15;          // 0..15
  const int rB = tid >> 4;          // 0..15

  float4 a_reg[4], b_reg[2];

  auto load_tile = [&](int kt) {
    #pragma unroll
    for (int p = 0; p < 4; ++p)
      a_reg[p] = *(const float4*)(A + (size_t)(m0 + rA + p * 32) * K + kt + caA);
    #pragma unroll
    for (int p = 0; p < 2; ++p)
      b_reg[p] = *(const float4*)(B + (size_t)(kt + rB + p * 16) * N + n0 + cB * 4);
  };
  auto store_tile = [&](int buf) {
    #pragma unroll
    for (int p = 0; p < 4; ++p) {
      float* dst = &Alds[buf][(rA + p * 32) * ASTR + caA];   // 8B aligned
      ((float2*)dst)[0] = make_float2(a_reg[p].x, a_reg[p].y);
      ((float2*)dst)[1] = make_float2(a_reg[p].z, a_reg[p].w);
    }
    #pragma unroll
    for (int p = 0; p < 2; ++p) {       // transpose 4(N) x 1(K) -> B^T[n][k]
      const int kr = rB + p * 16;
      Blds[buf][(cB * 4 + 0) * BTSTR + kr] = b_reg[p].x;
      Blds[buf][(cB * 4 + 1) * BTSTR + kr] = b_reg[p].y;
      Blds[buf][(cB * 4 + 2) * BTSTR + kr] = b_reg[p].z;
      Blds[buf][(cB * 4 + 3) * BTSTR + kr] = b_reg[p].w;
    }
  };

  const int aoff = (wm * 32 + l16) * ASTR;
  const int boff = (wn * 32 + l16) * BTSTR;

  load_tile(0);
  store_tile(0);
  __syncthreads();

  const int ntiles = K / KBLK;
  for (int t = 0; t < ntiles; ++t) {
    if (t + 1 < ntiles) {
      load_tile((t + 1) * KBLK);            // global loads overlap the WMMAs below
      if (t + 2 < ntiles) {                 // warm L2 for the tile after next
        __builtin_prefetch(A + (size_t)(m0 + rA) * K + (t + 2) * KBLK, 0, 1);
        __builtin_prefetch(B + (size_t)((t + 2) * KBLK + rB) * N + n0, 0, 1);
      }
    }
    const float* __restrict__ Ab = Alds[t & 1];
    const float* __restrict__ Bb = Blds[t & 1];
    #pragma unroll
    for (int ks = 0; ks < KBLK; ks += 4) {
      const int kb = ks + 2 * half;         // this lane's K-pair base
      v2f a0 = *(const v2f*)&Ab[aoff + kb];
      v2f a1 = *(const v2f*)&Ab[aoff + 16 * ASTR + kb];
      v2f b0 = *(const v2f*)&Bb[boff + kb];
      v2f b1 = *(const v2f*)&Bb[boff + 16 * BTSTR + kb];
      WMMA_F32(acc00, a0, b0);
      WMMA_F32(acc01, a0, b1);
      WMMA_F32(acc10, a1, b0);
      WMMA_F32(acc11, a1, b1);
    }
    if (t + 1 < ntiles) store_tile((t + 1) & 1);
    __syncthreads();
  }

  // ---- write back: accum VGPR i holds (row = i + 8*half, col = l16) ----
  const int rb = m0 + wm * 32 + 8 * half;
  const int cb = n0 + wn * 32 + l16;
  const float bv0 = fuse ? bias[cb] : 0.0f;
  const float bv1 = fuse ? bias[cb + 16] : 0.0f;
  #pragma unroll
  for (int i = 0; i < 8; ++i) {
    float v00 = acc00[i], v01 = acc01[i];
    float v10 = acc10[i], v11 = acc11[i];
    if (fuse) {
      v00 = fmaxf(v00 + bv0, 0.0f);
      v01 = fmaxf(v01 + bv1, 0.0f);
      v10 = fmaxf(v10 + bv0, 0.0f);
      v11 = fmaxf(v11 + bv1, 0.0f);
    }
    const size_t r0 = (size_t)(rb + i);
    const size_t r1 = (size_t)(rb + 16 + i);
    C[r0 * N + cb]      = v00;
    C[r0 * N + cb + 16] = v01;
    C[r1 * N + cb]      = v10;
    C[r1 * N + cb + 16] = v11;
  }
}

// Column sums of H [rows, 512] -> gsum[512] via per-block partials + f32 atomics.
__launch_bounds__(256)
__global__ void colsum_kernel(const float* __restrict__ H, float* __restrict__ gsum,
                              int rows_per_block) {
  const int t = threadIdx.x;
  const int r0 = blockIdx.x * rows_per_block;
  float s0 = 0.0f, s1 = 0.0f;
  for (int r = r0; r < r0 + rows_per_block; ++r) {
    s0 += H[(size_t)r * 512 + t];
    s1 += H[(size_t)r * 512 + t + 256];
  }
  atomicAdd(&gsum[t], s0);
  atomicAdd(&gsum[t + 256], s1);
}

// Epilogue: SELU(mean), fc1, concat, attention gating, log_softmax -> out[10].
__launch_bounds__(256)
__global__ void head_kernel(const float* __restrict__ gsum,
                            const float* __restrict__ sub_fea,
                            const float* __restrict__ fc1_W,
                            const float* __restrict__ fc1_b,
                            const float* __restrict__ att_W,
                            const float* __restrict__ att_a,
                            const float* __restrict__ att_b,
                            float* __restrict__ out) {
  __shared__ float z[768];
  __shared__ float scores[8];
  __shared__ float alpha[8];
  __shared__ float outv[10];
  const int t = threadIdx.x;

  const float sc = 1.0507009873554805f;
  const float al = 1.6732632423543772f;
  for (int c = t; c < 512; c += 256) {
    const float m = gsum[c] * (1.0f / 8192.0f);
    z[c] = sc * (m > 0.0f ? m : al * (expf(m) - 1.0f));
  }
  {
    float s = fc1_b[t];
    for (int i = 0; i < 128; ++i) s += sub_fea[i] * fc1_W[i * 256 + t];
    z[512 + t] = s;
  }
  __syncthreads();

  if (t < 8) {
    float s = 0.0f;
    const float* a = att_a + t * 768;
    for (int i = 0; i < 768; ++i) s += z[i] * a[i];
    scores[t] = s;
  }
  __syncthreads();
  if (t == 0) {
    float m = scores[0];
    for (int h = 1; h < 8; ++h) m = fmaxf(m, scores[h]);
    float den = 0.0f;
    for (int h = 0; h < 8; ++h) { alpha[h] = expf(scores[h] - m); den += alpha[h]; }
    const float inv = 1.0f / den;
    for (int h = 0; h < 8; ++h) alpha[h] *= inv;
  }
  __syncthreads();

  if (t < 10) {
    float o = att_b[t];
    for (int h = 0; h < 8; ++h) {
      float ho = 0.0f;
      const float* W = att_W + (size_t)h * 768 * 10 + t;  // att_W[h][i][t]
      for (int i = 0; i < 768; ++i) ho += z[i] * W[i * 10];
      o += alpha[h] * ho;
    }
    outv[t] = o;
  }
  __syncthreads();
  if (t == 0) {
    float m = outv[0];
    for (int o = 1; o < 10; ++o) m = fmaxf(m, outv[o]);
    float den = 0.0f;
    for (int o = 0; o < 10; ++o) den += expf(outv[o] - m);
    const float lse = m + logf(den);
    for (int o = 0; o < 10; ++o) out[o] = outv[o] - lse;
  }
}

extern "C" void kernel_launch(void* const* d_in, const int* in_sizes, int n_in,
                              void* d_out, int out_size, void* d_ws, size_t ws_size,
                              hipStream_t stream) {
  const float* x       = (const float*)d_in[0];   // [8192,512]
  const float* adj     = (const float*)d_in[1];   // [8192,8192]
  const float* sub_fea = (const float*)d_in[2];   // [1,128]
  const float* gc1_W   = (const float*)d_in[3];   // [512,256]
  const float* gc1_b   = (const float*)d_in[4];   // [256]
  const float* gc2_W   = (const float*)d_in[5];   // [256,512]
  const float* gc2_b   = (const float*)d_in[6];   // [512]
  const float* fc1_W   = (const float*)d_in[7];   // [128,256]
  const float* fc1_b   = (const float*)d_in[8];   // [256]
  const float* att_W   = (const float*)d_in[9];   // [8,768,10]
  const float* att_a   = (const float*)d_in[10];  // [8,768]
  const float* att_b   = (const float*)d_in[11];  // [10]
  float* out = (float*)d_out;

  char* ws = (char*)d_ws;
  float* XW1  = (float*)(ws);                           // 8192*256*4  =  8 MB
  float* H1   = (float*)(ws + ((size_t)8  << 20));      // 8192*256*4  =  8 MB
  float* HW2  = (float*)(ws + ((size_t)16 << 20));      // 8192*512*4  = 16 MB
  float* H2   = (float*)(ws + ((size_t)32 << 20));      // 8192*512*4  = 16 MB
  float* gsum = (float*)(ws + ((size_t)48 << 20));      // 512*4

  const int M = 8192;
  dim3 blk(256);

  // 1) XW1 = x @ gc1_W                       [8192,512]x[512,256]
  gemm_wmma_f32<<<dim3(256 / NBLK, M / MBLK), blk, 0, stream>>>(
      x, gc1_W, nullptr, XW1, M, 256, 512, 0);
  // 2) H1 = relu(adj @ XW1 + gc1_b)          [8192,8192]x[8192,256]
  gemm_wmma_f32<<<dim3(256 / NBLK, M / MBLK), blk, 0, stream>>>(
      adj, XW1, gc1_b, H1, M, 256, 8192, 1);
  // 3) HW2 = H1 @ gc2_W                      [8192,256]x[256,512]
  gemm_wmma_f32<<<dim3(512 / NBLK, M / MBLK), blk, 0, stream>>>(
      H1, gc2_W, nullptr, HW2, M, 512, 256, 0);
  // 4) H2 = relu(adj @ HW2 + gc2_b)          [8192,8192]x[8192,512]
  gemm_wmma_f32<<<dim3(512 / NBLK, M / MBLK), blk, 0, stream>>>(
      adj, HW2, gc2_b, H2, M, 512, 8192, 1);

  // 5) column sums of H2 -> gsum
  hipMemsetAsync(gsum, 0, 512 * sizeof(float), stream);
  colsum_kernel<<<dim3(32), blk, 0, stream>>>(H2, gsum, M / 32);

  // 6) pooled SELU + fc1 + attention head + log_softmax
  head_kernel<<<dim3(1), blk, 0, stream>>>(gsum, sub_fea, fc1_W, fc1_b,
                                           att_W, att_a, att_b, out);
}